// Conv2d_9457517986573
// MI455X (gfx1250) — compile-verified
//
#include <hip/hip_runtime.h>
#include <hip/hip_bf16.h>

typedef __attribute__((ext_vector_type(2))) float v2f;
typedef __attribute__((ext_vector_type(8))) float v8f;

#define IN_H     224
#define IN_W     224
#define OUT_HW   218
#define OC       64
#define KPAD     8                 // taps per filter row, padded 7 -> 8 (slot 7 = 0 weight)
#define WSTRIDE  58                // LDS weight row stride in floats (bank-conflict-free b64)
#define NTILE_W  14                // ceil(218/16)
#define TOTAL_IN (32 * IN_H * IN_W)
#define BASE_MAX (TOTAL_IN - 1351) // max col offset used = 6*224 + 6 = 1350

__global__ __launch_bounds__(256) void conv7x7_wmma_f32(
    const float* __restrict__ in, const float* __restrict__ wgt,
    float* __restrict__ out)
{
    __shared__ __align__(16) float wlds[OC * WSTRIDE];

    const int tid = threadIdx.x;
    // Cooperative weight image build: wlds[m*58 + kh*8 + kw]; kw==7 and tail -> 0.
    for (int i = tid; i < OC * WSTRIDE; i += 256) {
        int m  = i / WSTRIDE;
        int r  = i - m * WSTRIDE;
        int kh = r >> 3, kw = r & 7;
        float v = 0.0f;
        if (r < 56 && kw < 7) v = wgt[m * 49 + kh * 7 + kw];
        wlds[i] = v;
    }
    __syncthreads();

    const int lane   = tid & 31;
    const int lane16 = lane & 15;
    const int hi     = lane >> 4;      // 0: lanes 0-15 (k-slots 0,1), 1: lanes 16-31 (k-slots 2,3)
    const int co     = hi * 2;

    const int wid = blockIdx.x * 8 + (tid >> 5);
    const int tw  = wid % NTILE_W;
    const int rem = wid / NTILE_W;
    const int oh  = rem % OUT_HW;
    const int n   = rem / OUT_HW;

    const int ow0 = tw * 16;
    int base = (n * IN_H + oh) * IN_W + ow0 + lane16;
    if (base > BASE_MAX) base = BASE_MAX;   // only clamps discarded edge lanes
    const float* p = in + base;

    const float* arow = &wlds[lane16 * WSTRIDE + co];

    v8f acc0 = {}, acc1 = {}, acc2 = {}, acc3 = {};

#pragma unroll
    for (int kh = 0; kh < 7; ++kh) {
        const float* pr = p + kh * IN_W;
        const int kb = kh * KPAD;

        // ---- K-chunk 0: taps {0,1,2,3} of this filter row ----
        v2f b;
        b.x = pr[co];
        b.y = pr[co + 1];
        {
            v2f a0 = *(const v2f*)(arow + kb);
            v2f a1 = *(const v2f*)(arow + kb + 16 * WSTRIDE);
            v2f a2 = *(const v2f*)(arow + kb + 32 * WSTRIDE);
            v2f a3 = *(const v2f*)(arow + kb + 48 * WSTRIDE);
            acc0 = __builtin_amdgcn_wmma_f32_16x16x4_f32(false, a0, false, b, (short)0, acc0, false, false);
            acc1 = __builtin_amdgcn_wmma_f32_16x16x4_f32(false, a1, false, b, (short)0, acc1, false, false);
            acc2 = __builtin_amdgcn_wmma_f32_16x16x4_f32(false, a2, false, b, (short)0, acc2, false, false);
            acc3 = __builtin_amdgcn_wmma_f32_16x16x4_f32(false, a3, false, b, (short)0, acc3, false, false);
        }

        // ---- K-chunk 1: taps {4,5,6, pad}; pad slot has zero weight so the
        //      dummy col-6 load in lanes>=16 contributes exactly 0 ----
        b.x = pr[co + 4];
        b.y = pr[hi ? 6 : 5];
        {
            v2f a0 = *(const v2f*)(arow + kb + 4);
            v2f a1 = *(const v2f*)(arow + kb + 4 + 16 * WSTRIDE);
            v2f a2 = *(const v2f*)(arow + kb + 4 + 32 * WSTRIDE);
            v2f a3 = *(const v2f*)(arow + kb + 4 + 48 * WSTRIDE);
            acc0 = __builtin_amdgcn_wmma_f32_16x16x4_f32(false, a0, false, b, (short)0, acc0, false, false);
            acc1 = __builtin_amdgcn_wmma_f32_16x16x4_f32(false, a1, false, b, (short)0, acc1, false, false);
            acc2 = __builtin_amdgcn_wmma_f32_16x16x4_f32(false, a2, false, b, (short)0, acc2, false, false);
            acc3 = __builtin_amdgcn_wmma_f32_16x16x4_f32(false, a3, false, b, (short)0, acc3, false, false);
        }
    }

    // ---- Store 64 channels x 16 columns (EXEC divergence is fine post-WMMA) ----
    const int ow = ow0 + lane16;
    if (ow < OUT_HW) {
        const size_t plane = (size_t)OUT_HW * OUT_HW;
        float* o = out + (((size_t)n * OC + hi * 8) * OUT_HW + oh) * OUT_HW + ow;
#pragma unroll
        for (int j = 0; j < 8; ++j) {
            o[(size_t)(j)      * plane] = acc0[j];
            o[(size_t)(16 + j) * plane] = acc1[j];
            o[(size_t)(32 + j) * plane] = acc2[j];
            o[(size_t)(48 + j) * plane] = acc3[j];
        }
    }
}

extern "C" void kernel_launch(void* const* d_in, const int* in_sizes, int n_in,
                              void* d_out, int out_size, void* d_ws, size_t ws_size,
                              hipStream_t stream) {
    const float* data = (const float*)d_in[0];   // 32*1*224*224 f32
    const float* wgt  = (const float*)d_in[1];   // 64*1*7*7 f32
    float* out = (float*)d_out;                  // 32*64*218*218 f32

    // 32 * 218 * 14 wave-tiles, 8 waves (256 threads) per block -> exactly 12208 blocks
    dim3 grid(12208), block(256);
    hipLaunchKernelGGL(conv7x7_wmma_f32, grid, block, 0, stream, data, wgt, out);
}